// update_v_87840671137924
// MI455X (gfx1250) — compile-verified
//
#include <hip/hip_runtime.h>

typedef __attribute__((ext_vector_type(2))) float v2f;
typedef __attribute__((ext_vector_type(8))) float v8f;

#define HDIM 128
#define LDSS 132   // 128 + 4 pad: 16B-aligned rows, conflict-free column reads

// ---------------- zero the aggregation workspace ----------------
__global__ __launch_bounds__(256) void zero_f4(float4* __restrict__ p, int n4) {
    int i = blockIdx.x * 256 + threadIdx.x;
    int stride = gridDim.x * 256;
    float4 z = {0.f, 0.f, 0.f, 0.f};
    for (; i < n4; i += stride) p[i] = z;
}

// ---------------- scatter-add edge messages into nodes ----------------
// 32 lanes per edge, 4 floats per lane (float4 coalesced edge read).
__global__ __launch_bounds__(256) void scatter_add(const float* __restrict__ e,
                                                   const long long* __restrict__ tgt,
                                                   float* __restrict__ agg, int E) {
    int t = blockIdx.x * 256 + threadIdx.x;
    int j = t >> 5;
    if (j >= E) return;
    int lane = t & 31;
    long long node = tgt[j];
    const float4 val = *(const float4*)(e + (size_t)j * HDIM + lane * 4);
    float* dst = agg + (size_t)node * HDIM + lane * 4;
    atomicAdd(dst + 0, val.x);
    atomicAdd(dst + 1, val.y);
    atomicAdd(dst + 2, val.z);
    atomicAdd(dst + 3, val.w);
}

// branchless, numerically stable shifted softplus: max(x,0)+log1p(exp(-|x|)) - ln2
__device__ __forceinline__ float sspe(float x) {
    float t = __expf(-__builtin_fabsf(x));
    return fmaxf(x, 0.f) + __logf(1.f + t) - 0.69314718055994531f;
}

// ---------------- fused MLP: out = v + (sspe(agg@w1+b1)) @ w2 + b2 ----------
// One block = 16 output rows x 128 cols. 8 waves; wave w owns cols [16w,16w+16).
// f32 WMMA 16x16x4, K-loop of 32 steps per layer.
__global__ __launch_bounds__(256) void fused_mlp(const float* __restrict__ agg,
                                                 const float* __restrict__ v,
                                                 const float* __restrict__ w1,
                                                 const float* __restrict__ b1,
                                                 const float* __restrict__ w2,
                                                 const float* __restrict__ b2,
                                                 float* __restrict__ out) {
    __shared__ float lds_a[16 * LDSS];   // staged agg strip (layer-1 A operand)
    __shared__ float lds_t[16 * LDSS];   // activation strip (layer-2 A operand)

    const int tid  = threadIdx.x;
    const int row0 = blockIdx.x * 16;

    // Stage agg[row0:row0+16, :] into LDS, float4-coalesced.
    for (int idx = tid; idx < 16 * 32; idx += 256) {
        int r = idx >> 5, c4 = idx & 31;
        float4 d = *(const float4*)(agg + (size_t)(row0 + r) * HDIM + c4 * 4);
        *(float4*)(lds_a + r * LDSS + c4 * 4) = d;
    }
    __syncthreads();

    const int wave = tid >> 5;
    const int lane = tid & 31;
    const int half = lane >> 4;        // 0: lanes 0-15, 1: lanes 16-31
    const int l16  = lane & 15;
    const int n    = wave * 16 + l16;  // output column owned by this lane
    const float bias1 = b1[n];
    const float bias2 = b2[n];

    // ---- layer 1: acc = agg_strip @ w1[:, n-tile] ----
    // A 16x4 f32 layout: lanes 0-15 hold M=l16, K={kk,kk+1}; lanes 16-31 K={kk+2,kk+3}
    // B 4x16 f32 layout: VGPR0 = row (kk+2*half), VGPR1 = row (kk+2*half+1), N=l16
    v8f acc = {};
    #pragma unroll 8
    for (int kk = 0; kk < HDIM; kk += 4) {
        const int ka = kk + half * 2;
        v2f a;
        a.x = lds_a[l16 * LDSS + ka];
        a.y = lds_a[l16 * LDSS + ka + 1];
        v2f b;
        b.x = w1[(size_t)ka * HDIM + n];
        b.y = w1[(size_t)(ka + 1) * HDIM + n];
        acc = __builtin_amdgcn_wmma_f32_16x16x4_f32(false, a, false, b,
                                                    (short)0, acc, false, false);
    }

    // shifted softplus (branchless), write activation strip to LDS
    // C/D layout: VGPR r -> M = r + 8*half, N = n
    #pragma unroll
    for (int r = 0; r < 8; ++r) {
        lds_t[(r + half * 8) * LDSS + n] = sspe(acc[r] + bias1);
    }
    __syncthreads();

    // ---- layer 2: acc2 = t_strip @ w2[:, n-tile] ----
    v8f acc2 = {};
    #pragma unroll 8
    for (int kk = 0; kk < HDIM; kk += 4) {
        const int ka = kk + half * 2;
        v2f a;
        a.x = lds_t[l16 * LDSS + ka];
        a.y = lds_t[l16 * LDSS + ka + 1];
        v2f b;
        b.x = w2[(size_t)ka * HDIM + n];
        b.y = w2[(size_t)(ka + 1) * HDIM + n];
        acc2 = __builtin_amdgcn_wmma_f32_16x16x4_f32(false, a, false, b,
                                                     (short)0, acc2, false, false);
    }

    // residual + bias, store
    #pragma unroll
    for (int r = 0; r < 8; ++r) {
        const int m = r + half * 8;
        const size_t off = (size_t)(row0 + m) * HDIM + n;
        out[off] = v[off] + acc2[r] + bias2;
    }
}

extern "C" void kernel_launch(void* const* d_in, const int* in_sizes, int n_in,
                              void* d_out, int out_size, void* d_ws, size_t ws_size,
                              hipStream_t stream) {
    const float*     v   = (const float*)d_in[0];
    const float*     e   = (const float*)d_in[1];
    const long long* ei  = (const long long*)d_in[2];  // int64 in reference, shape (2,E)
    const float*     w1  = (const float*)d_in[3];
    const float*     b1  = (const float*)d_in[4];
    const float*     w2  = (const float*)d_in[5];
    const float*     b2  = (const float*)d_in[6];
    float*           out = (float*)d_out;
    float*           agg = (float*)d_ws;               // N*HDIM f32 scratch

    const int N = in_sizes[0] / HDIM;   // 50000
    const int E = in_sizes[1] / HDIM;   // 600000

    // 1) zero agg
    const int n4 = (N * HDIM) / 4;
    zero_f4<<<dim3((n4 + 255) / 256), dim3(256), 0, stream>>>((float4*)agg, n4);

    // 2) scatter-add edges into agg (target node row = edge_index[1] at offset E)
    const long long total = (long long)E * 32;
    const int sblocks = (int)((total + 255) / 256);
    scatter_add<<<dim3(sblocks), dim3(256), 0, stream>>>(e, ei + E, agg, E);

    // 3) fused two-layer MLP + residual (N assumed multiple of 16: 50000 = 3125*16)
    fused_mlp<<<dim3(N / 16), dim3(256), 0, stream>>>(agg, v, w1, b1, w2, b2, out);
}